// Conv2d_NN_sanity_32246614458953
// MI455X (gfx1250) — compile-verified
//
#include <hip/hip_runtime.h>

// Problem constants (from reference): B=4, C_IN=32, C_OUT=64, H=W=64, K=9, SIGMA=0.1
#define T_PIX 4096           // H*W
#define NB 4                 // batch
#define CIN 32
#define COUT 64
#define KNN 9
#define KRED 288             // CIN*KNN (GEMM reduction dim)
#define LDS_STRIDE 290       // padded A-tile row stride (even, 290%64=34 -> conflict-free)

typedef float v2f __attribute__((ext_vector_type(2)));
typedef float v8f __attribute__((ext_vector_type(8)));

// ---------------------------------------------------------------------------
// Kernel 1: per-pixel top-9 of the Gaussian similarity (input-independent).
//
// sim = exp(-(max(d2,0)+1e-8)/0.02) is monotone-decreasing in Euclidean
// distance, so the 9 largest values must come from the clamped 5x5 window:
// every pixel has >=9 window candidates at distance <= sqrt(8)*h while any
// pixel outside the window is at distance >= 3*h, and exp(-50*d2) cannot tie
// across that gap in fp32. Scanning the window in ascending flat index with
// strict '>' insertion reproduces jax.lax.top_k exactly (descending value,
// ties -> lower index), since ties only occur between window members.
// ---------------------------------------------------------------------------
__global__ __launch_bounds__(256)
void topk_idx_kernel(int* __restrict__ idx_out) {
  const int t = blockIdx.x * 256 + threadIdx.x;       // grid is exactly T_PIX wide
  const float step = 2.0f / 63.0f;                    // linspace(-1,1,64) spacing
  const int ri = t >> 6, ci = t & 63;
  const float yi = -1.0f + step * (float)ri;
  const float xi = -1.0f + step * (float)ci;
  const float sqi = xi * xi + yi * yi;

  float val[KNN];
  int   ind[KNN];
#pragma unroll
  for (int k = 0; k < KNN; ++k) { val[k] = -1.0f; ind[k] = 0; }

  const int r0 = (ri - 2 < 0) ? 0 : ri - 2;
  const int r1 = (ri + 2 > 63) ? 63 : ri + 2;
  const int c0 = (ci - 2 < 0) ? 0 : ci - 2;
  const int c1 = (ci + 2 > 63) ? 63 : ci + 2;

  for (int rj = r0; rj <= r1; ++rj) {                 // ascending rows ...
    const float yj = -1.0f + step * (float)rj;
    for (int cj = c0; cj <= c1; ++cj) {               // ... then cols => j ascending
      const int j = (rj << 6) + cj;
      const float xj = -1.0f + step * (float)cj;
      const float sqj = xj * xj + yj * yj;
      float d2 = sqi + sqj - 2.0f * (xi * xj + yi * yj);
      float dd = fmaxf(d2, 0.0f) + 1e-8f;             // dist*dist
      float sim = expf(-dd / 0.02f);                  // exp(-d^2 / (2*sigma^2))

      if (sim > val[KNN - 1]) {                       // beats current 9th?
        // Unrolled stable insertion: constant indices -> stays in VGPRs.
#pragma unroll
        for (int p = KNN - 1; p > 0; --p) {
          bool shift = sim > val[p - 1];              // slot receives shifted value
          bool here  = (!shift) && (sim > val[p]);    // unique insertion slot
          val[p] = shift ? val[p - 1] : (here ? sim : val[p]);
          ind[p] = shift ? ind[p - 1] : (here ? j   : ind[p]);
        }
        if (sim > val[0]) { val[0] = sim; ind[0] = j; }
      }
    }
  }
#pragma unroll
  for (int k = 0; k < KNN; ++k) idx_out[t * KNN + k] = ind[k];
}

// ---------------------------------------------------------------------------
// Kernel 2: gather-GEMM, D[M=16, N=16] per wave via V_WMMA_F32_16X16X4_F32.
//   out[b,o,t] = sum_K A[m,K] * Bmat[K,o] + bias[o]
//   A[m,K]    = x[b, K/9, idx[t, K%9]]   (gathered into LDS per block)
//   Bmat[K,o] = conv_w flat at o*288+K   (row-major (O,I,K) == N-major over K)
// Block = 128 threads (4 waves): one 16-row M-tile, waves split N=64 into 4x16.
// ---------------------------------------------------------------------------
__global__ __launch_bounds__(128)
void gemm_wmma_kernel(const float* __restrict__ x,
                      const float* __restrict__ w,
                      const float* __restrict__ bias,
                      const int*   __restrict__ idx,
                      float*       __restrict__ out) {
  __shared__ __align__(16) float sA[16 * LDS_STRIDE];
  __shared__ int sIdx[16 * KNN];

  const int tb  = blockIdx.x;         // t-tile: 0..255
  const int b   = blockIdx.y;         // batch:  0..3
  const int t0  = tb * 16;
  const int tid = threadIdx.x;

  // Stage the 16x9 neighbor-index block (contiguous in idx).
  if (tid < 16 * KNN) sIdx[tid] = idx[t0 * KNN + tid];
  __syncthreads();

  // Cooperative gather of the 16x288 A-tile into padded LDS.
  const float* xb = x + (size_t)b * (CIN * T_PIX);
  for (int e = tid; e < 16 * KRED; e += 128) {
    int r  = e / KRED;
    int K  = e - r * KRED;
    int c  = K / KNN;
    int kk = K - c * KNN;
    int sp = sIdx[r * KNN + kk];
    sA[r * LDS_STRIDE + K] = xb[c * T_PIX + sp];
  }
  __syncthreads();

  const int lane = tid & 31;
  const int wv   = tid >> 5;
  const int n0   = wv * 16;
  const int rr   = lane & 15;         // A row (M) for lanes; B/D column (N)
  const int hi   = lane >> 4;         // half-wave: K-pair select / D row offset
  const int n    = n0 + rr;           // output channel this lane owns for B/D

  const float* wrow = w + n * KRED;                  // Bmat[:, n] == w[n*288 + K]
  const float* aRow = &sA[rr * LDS_STRIDE];

  v8f acc = {};
#pragma unroll 4
  for (int s = 0; s < KRED / 4; ++s) {
    const int kb = 4 * s + 2 * hi;                   // even -> 8B-aligned b64 loads
    v2f a  = *(const v2f*)(aRow + kb);               // A frag: M=rr, K=kb..kb+1
    v2f bf = *(const v2f*)(wrow + kb);               // B frag: K=kb..kb+1, N=n
    acc = __builtin_amdgcn_wmma_f32_16x16x4_f32(
        /*neg_a=*/false, a, /*neg_b=*/false, bf,
        /*c_mod=*/(short)0, acc, /*reuse_a=*/false, /*reuse_b=*/false);
  }

  // D layout: VGPR v -> row v (lanes 0-15) / row v+8 (lanes 16-31), N = rr.
  const float bv = bias[n];
  float* outp = out + ((size_t)b * COUT + n) * T_PIX + t0;
#pragma unroll
  for (int v = 0; v < 8; ++v) {
    outp[v + 8 * hi] = acc[v] + bv;
  }
}

// ---------------------------------------------------------------------------
extern "C" void kernel_launch(void* const* d_in, const int* in_sizes, int n_in,
                              void* d_out, int out_size, void* d_ws, size_t ws_size,
                              hipStream_t stream) {
  const float* x    = (const float*)d_in[0];   // (4,32,64,64) f32
  const float* w    = (const float*)d_in[1];   // (64,32,9)    f32
  const float* bias = (const float*)d_in[2];   // (64,)        f32
  int* idx_ws = (int*)d_ws;                    // 4096*9 ints (147 KB) scratch

  topk_idx_kernel<<<dim3(T_PIX / 256), 256, 0, stream>>>(idx_ws);
  gemm_wmma_kernel<<<dim3(T_PIX / 16, NB), 128, 0, stream>>>(
      x, w, bias, idx_ws, (float*)d_out);
}